// GraphConv_34179349742034
// MI455X (gfx1250) — compile-verified
//
#include <hip/hip_runtime.h>
#include <hip/hip_bf16.h>

// GraphConv (norm='both'), D_FEAT = 64 fixed.
// Memory-bound gather/scatter; whole working set (~64 MB) fits MI455X's 192 MB L2.
// CDNA5 paths used: GLOBAL_LOAD_ASYNC_TO_LDS_B32 (+ s_wait_asynccnt) for edge
// staging, ds_load broadcast reads, non-returning global_atomic_add_f32.

#define DFEAT 64
#define WAVESZ 32
#define BLK 256
#define WAVES_PER_BLK (BLK / WAVESZ)

__global__ __launch_bounds__(BLK)
void gc_init_kernel(float* __restrict__ out, unsigned* __restrict__ outdeg,
                    unsigned* __restrict__ indeg, int n_nodes, int total) {
    int i = blockIdx.x * BLK + threadIdx.x;
    if (i < total) out[i] = 0.0f;
    if (i < n_nodes) { outdeg[i] = 0u; indeg[i] = 0u; }
}

__global__ __launch_bounds__(BLK)
void gc_degree_kernel(const int* __restrict__ src, const int* __restrict__ dst,
                      unsigned* __restrict__ outdeg, unsigned* __restrict__ indeg, int E) {
    int e = blockIdx.x * BLK + threadIdx.x;
    if (e < E) {
        atomicAdd(&outdeg[src[e]], 1u);
        atomicAdd(&indeg[dst[e]], 1u);
    }
}

__global__ __launch_bounds__(BLK)
void gc_norm_kernel(const unsigned* __restrict__ outdeg, const unsigned* __restrict__ indeg,
                    float* __restrict__ onorm, float* __restrict__ inorm, int n_nodes) {
    int i = blockIdx.x * BLK + threadIdx.x;
    if (i < n_nodes) {
        onorm[i] = 1.0f / sqrtf(fmaxf((float)outdeg[i], 1.0f));
        inorm[i] = 1.0f / sqrtf(fmaxf((float)indeg[i], 1.0f));
    }
}

// One wave handles a chunk of 32 edges. Edge indices are staged into LDS via
// CDNA5 async global->LDS loads (per-lane), then each edge is processed
// cooperatively: 32 lanes x float2 covers the 64-float feature row.
__global__ __launch_bounds__(BLK)
void gc_aggregate_kernel(const float* __restrict__ feat,
                         const int* __restrict__ src, const int* __restrict__ dst,
                         const float* __restrict__ onorm,
                         float* __restrict__ out, int E) {
    __shared__ int s_src[WAVES_PER_BLK * WAVESZ];
    __shared__ int s_dst[WAVES_PER_BLK * WAVESZ];

    const int lane = threadIdx.x & (WAVESZ - 1);
    const int wid  = threadIdx.x >> 5;
    const int gwave = (blockIdx.x * BLK + threadIdx.x) >> 5;
    const int nwaves = (gridDim.x * BLK) >> 5;

    // Per-lane LDS destination byte offsets (low 32 bits of the generic
    // address of a __shared__ object == wave-relative LDS byte offset).
    const unsigned lds_src = (unsigned)(unsigned long long)(void*)&s_src[wid * WAVESZ + lane];
    const unsigned lds_dst = (unsigned)(unsigned long long)(void*)&s_dst[wid * WAVESZ + lane];
    const int wbase = wid * WAVESZ;

    for (int chunk = gwave * WAVESZ; chunk < E; chunk += nwaves * WAVESZ) {
        int e = chunk + lane;
        if (e >= E) e = E - 1;  // clamp: safe load, skipped in j-loop

        // Async-stage this wave's 32 src/dst indices into LDS (ASYNCcnt path).
        asm volatile("global_load_async_to_lds_b32 %0, %1, off"
                     :: "v"(lds_src), "v"((unsigned long long)(const void*)(src + e))
                     : "memory");
        asm volatile("global_load_async_to_lds_b32 %0, %1, off"
                     :: "v"(lds_dst), "v"((unsigned long long)(const void*)(dst + e))
                     : "memory");
        asm volatile("s_wait_asynccnt 0" ::: "memory");

        int limit = E - chunk;
        if (limit > WAVESZ) limit = WAVESZ;

        for (int j = 0; j < limit; ++j) {
            const int sj = s_src[wbase + j];   // uniform ds_load -> broadcast
            const int dj = s_dst[wbase + j];
            const float w = onorm[sj];          // uniform address -> 1 L2 txn/wave

            // 32 lanes x float2 = full 64-float row, 256 B coalesced (L2-resident).
            const float2 v = *(const float2*)(feat + (size_t)sj * DFEAT + lane * 2);
            float* o = out + (size_t)dj * DFEAT + lane * 2;
            atomicAdd(o,     v.x * w);          // non-returning global_atomic_add_f32
            atomicAdd(o + 1, v.y * w);
        }
    }
}

__global__ __launch_bounds__(BLK)
void gc_scale_kernel(float* __restrict__ out, const float* __restrict__ inorm, int total) {
    int i = blockIdx.x * BLK + threadIdx.x;
    if (i < total) out[i] *= inorm[i >> 6];  // row = i / 64
}

extern "C" void kernel_launch(void* const* d_in, const int* in_sizes, int n_in,
                              void* d_out, int out_size, void* d_ws, size_t ws_size,
                              hipStream_t stream) {
    const float* feat = (const float*)d_in[0];
    const int*   src  = (const int*)d_in[1];
    const int*   dst  = (const int*)d_in[2];
    float* out = (float*)d_out;

    const int n_nodes = in_sizes[0] / DFEAT;
    const int E       = in_sizes[1];
    const int total   = n_nodes * DFEAT;

    // Workspace layout: outdeg | indeg (u32), onorm | inorm (f32): 16*N bytes.
    unsigned* outdeg = (unsigned*)d_ws;
    unsigned* indeg  = outdeg + n_nodes;
    float*    onorm  = (float*)(indeg + n_nodes);
    float*    inorm  = onorm + n_nodes;

    const int gInit  = (total + BLK - 1) / BLK;
    const int gEdge  = (E + BLK - 1) / BLK;
    const int gNode  = (n_nodes + BLK - 1) / BLK;
    // One wave per 32-edge chunk, 8 waves/block.
    int gAgg = ((E + WAVESZ - 1) / WAVESZ + WAVES_PER_BLK - 1) / WAVES_PER_BLK;
    if (gAgg < 1) gAgg = 1;

    gc_init_kernel<<<gInit, BLK, 0, stream>>>(out, outdeg, indeg, n_nodes, total);
    gc_degree_kernel<<<gEdge, BLK, 0, stream>>>(src, dst, outdeg, indeg, E);
    gc_norm_kernel<<<gNode, BLK, 0, stream>>>(outdeg, indeg, onorm, inorm, n_nodes);
    gc_aggregate_kernel<<<gAgg, BLK, 0, stream>>>(feat, src, dst, onorm, out, E);
    gc_scale_kernel<<<gInit, BLK, 0, stream>>>(out, inorm, total);
}